// AspectMemoryEncoder_24472723652705
// MI455X (gfx1250) — compile-verified
//
#include <hip/hip_runtime.h>
#include <hip/hip_bf16.h>

typedef __attribute__((ext_vector_type(2))) float v2f;
typedef __attribute__((ext_vector_type(8))) float v8f;

#define B_DIM 1024
#define S_DIM 256
#define E_DIM 200
#define A_DIM 14
#define EPS   1e-8f

// ---------------------------------------------------------------------------
// Kernel 1: normalize aspect embeddings into a 16 x 200 padded matrix.
// Rows 14,15 are COPIES of row 0 so that max over all 16 rows == max over 14.
// ---------------------------------------------------------------------------
__global__ __launch_bounds__(256) void aspect_norm_kernel(const float* __restrict__ a_emb,
                                                          float* __restrict__ a_n) {
    __shared__ float norms[16];
    int t = threadIdx.x;
    if (t < A_DIM) {
        float ss = 0.f;
        for (int e = 0; e < E_DIM; ++e) {
            float v = a_emb[t * E_DIM + e];
            ss = fmaf(v, v, ss);
        }
        norms[t] = fmaxf(sqrtf(ss), EPS);
    }
    __syncthreads();
    for (int i = t; i < 16 * E_DIM; i += 256) {
        int r = i / E_DIM;
        int e = i - r * E_DIM;
        int src = (r < A_DIM) ? r : 0;             // pad rows 14,15 with row 0
        a_n[i] = a_emb[src * E_DIM + e] / norms[src];
    }
}

// ---------------------------------------------------------------------------
// Kernel 2: sim[b,s] = max_a cos(a_emb[a], w_emb[ids[b,s]])
// One wave32 per 16-token tile. D(16x16) = A_n(16x200) * X(200x16) via 50
// chained V_WMMA_F32_16X16X4_F32 (exact f32). Normalization factored out:
// divide the row-max by max(||x||, eps) afterwards.
//
// f32 16x16x4 operand layout (ISA 7.12.2):
//   A: lane m (0-15)  -> {A[m][4k+0], A[m][4k+1]},  lane m+16 -> {A[m][4k+2], A[m][4k+3]}
//   B: lane n (0-15)  -> {B[4k+0][n], B[4k+1][n]},  lane n+16 -> {B[4k+2][n], B[4k+3][n]}
// So each lane loads float2 at element offset 4k + 2*(lane>>4) from its row.
// ---------------------------------------------------------------------------
__global__ __launch_bounds__(128) void sim_wmma_kernel(const int* __restrict__ ids,
                                                       const float* __restrict__ w_emb,
                                                       const float* __restrict__ a_n,
                                                       float* __restrict__ sim) {
    __shared__ float lds_an[16 * E_DIM];
    int t = threadIdx.x;
    for (int i = t; i < 16 * E_DIM; i += 128) lds_an[i] = a_n[i];
    __syncthreads();

    const int lane = t & 31;
    const int wave = t >> 5;
    const int tile = blockIdx.x * 4 + wave;     // 16 tokens per wave
    const int col  = lane & 15;                 // aspect row for A / token column for B
    const int half = lane >> 4;                 // 0: K%4 in {0,1}, 1: K%4 in {2,3}
    const int token = tile * 16 + col;

    const float* __restrict__ xrow = w_emb + (size_t)ids[token] * E_DIM;
    const float* __restrict__ arow = lds_an + col * E_DIM;

    v8f acc = {0.f, 0.f, 0.f, 0.f, 0.f, 0.f, 0.f, 0.f};
    float ss = 0.f;

    #pragma unroll 10
    for (int k = 0; k < E_DIM / 4; ++k) {       // 50 K-chunks of 4
        int off = 4 * k + 2 * half;
        v2f bv = *(const v2f*)(xrow + off);     // token data (global, L2-resident)
        v2f av = *(const v2f*)(arow + off);     // aspect data (LDS)
        ss = fmaf(bv.x, bv.x, ss);
        ss = fmaf(bv.y, bv.y, ss);
        acc = __builtin_amdgcn_wmma_f32_16x16x4_f32(
            /*neg_a=*/false, av, /*neg_b=*/false, bv,
            /*c_mod=*/(short)0, acc, /*reuse_a=*/false, /*reuse_b=*/false);
    }

    // Row-max over this lane's 8 rows (lanes 0-15: rows 0-7, lanes 16-31: rows 8-15).
    float m = acc[0];
    m = fmaxf(m, acc[1]); m = fmaxf(m, acc[2]); m = fmaxf(m, acc[3]);
    m = fmaxf(m, acc[4]); m = fmaxf(m, acc[5]); m = fmaxf(m, acc[6]);
    m = fmaxf(m, acc[7]);
    // Merge the two half-wave partners (wave32 shuffle).
    m  = fmaxf(m, __shfl_xor(m, 16));
    float ss_all = ss + __shfl_xor(ss, 16);
    float nrm = fmaxf(sqrtf(ss_all), EPS);
    if (half == 0) sim[token] = m / nrm;
}

// ---------------------------------------------------------------------------
// Kernel 3: per batch row: softmax over S, then z[b,e] = sum_s a[b,s]*x[b,s,e].
// One 256-thread block per batch row; ids/weights staged in LDS; the 200
// active threads make the w_emb row reads coalesced (800B rows, L2-resident).
// ---------------------------------------------------------------------------
__global__ __launch_bounds__(256) void softmax_z_kernel(const int* __restrict__ ids,
                                                        const float* __restrict__ w_emb,
                                                        const float* __restrict__ sim,
                                                        float* __restrict__ z,
                                                        float* __restrict__ a_out) {
    __shared__ float s_a[S_DIM];
    __shared__ int   s_id[S_DIM];
    __shared__ float red[S_DIM];

    const int b = blockIdx.x;
    const int t = threadIdx.x;

    float v = sim[b * S_DIM + t];
    red[t] = v;
    __syncthreads();
    for (int s = S_DIM / 2; s > 0; s >>= 1) {
        if (t < s) red[t] = fmaxf(red[t], red[t + s]);
        __syncthreads();
    }
    float vmax = red[0];
    __syncthreads();

    float e = expf(v - vmax);
    red[t] = e;
    __syncthreads();
    for (int s = S_DIM / 2; s > 0; s >>= 1) {
        if (t < s) red[t] += red[t + s];
        __syncthreads();
    }
    float a = e / red[0];

    a_out[b * S_DIM + t] = a;
    s_a[t]  = a;
    s_id[t] = ids[b * S_DIM + t];
    __syncthreads();

    if (t < E_DIM) {
        float acc = 0.f;
        #pragma unroll 8
        for (int s = 0; s < S_DIM; ++s) {
            acc = fmaf(s_a[s], w_emb[(size_t)s_id[s] * E_DIM + t], acc);
        }
        z[b * E_DIM + t] = acc;
    }
}

// ---------------------------------------------------------------------------
// Launch
// ---------------------------------------------------------------------------
extern "C" void kernel_launch(void* const* d_in, const int* in_sizes, int n_in,
                              void* d_out, int out_size, void* d_ws, size_t ws_size,
                              hipStream_t stream) {
    const int*   ids   = (const int*)d_in[0];    // (B,S) token ids
    const float* w_emb = (const float*)d_in[1];  // (V,E)
    const float* a_emb = (const float*)d_in[2];  // (A,E)

    float* z     = (float*)d_out;                         // (B,E) = 204800 floats
    float* a_out = (float*)d_out + B_DIM * E_DIM;         // (B,S) = 262144 floats

    // Workspace: a_n (16x200 f32 = 12.8KB) at offset 0; sim (B*S f32 = 1MB) at 16KB.
    float* a_n = (float*)d_ws;
    float* sim = (float*)((char*)d_ws + 16384);

    aspect_norm_kernel<<<1, 256, 0, stream>>>(a_emb, a_n);

    const int n_tiles = (B_DIM * S_DIM) / 16;   // 16384 tiles, 4 per block
    sim_wmma_kernel<<<n_tiles / 4, 128, 0, stream>>>(ids, w_emb, a_n, sim);

    softmax_z_kernel<<<B_DIM, 256, 0, stream>>>(ids, w_emb, sim, z, a_out);
}